// GramLoss_31413390802979
// MI455X (gfx1250) — compile-verified
//
#include <hip/hip_runtime.h>
#include <hip/hip_bf16.h>
#include <math.h>

typedef __attribute__((ext_vector_type(2))) float v2f;
typedef __attribute__((ext_vector_type(8))) float v8f;

#define NB    4
#define IMG   72
#define N1    576      // 24*24 patches per image
#define MCAT  756      // 576 + 144 + 36
#define MPAD  768      // padded to 48 column tiles of 16
#define RSTR  16       // padded row stride of pcat: [0..8]=desc, 9..11=0, [12]=norm
#define TOTEL 20736.0f // B*N1*9

// ---------------- workspace layout (float units) ----------------
// p1    [4][576][9]        off 0       20736
// pcat  [4][768][16]       off 20736   49152
// g2    [4][3][36][36]     off 69888   15552
// g4    [4][3][18][18]     off 85440    3888
// idx   [4][576] (int)     off 89328    2304
// part  [9]                off 91632       9
#define OFF_P1    0
#define OFF_PCAT  20736
#define OFF_G2    69888
#define OFF_G4    85440
#define OFF_IDX   89328
#define OFF_PART  91632

// fixed bicubic taps for a=-0.75, align_corners=False, integer factor
__device__ __constant__ float cW4[4] = {-0.09375f, 0.59375f, 0.59375f, -0.09375f};

// ---------------- bicubic downsample, factor 2: 72 -> 36 ----------------
__global__ __launch_bounds__(256) void down2_kernel(const float* __restrict__ gt,
                                                    float* __restrict__ g2) {
    int t = blockIdx.x * 256 + threadIdx.x;
    if (t >= NB * 3 * 36 * 36) return;
    int bc = t / 1296;
    int p  = t % 1296;
    int oy = p / 36, ox = p % 36;
    const float* in = gt + (size_t)bc * (IMG * IMG);
    int iy0 = 2 * oy - 1, ix0 = 2 * ox - 1;
    float acc = 0.f;
    #pragma unroll
    for (int ky = 0; ky < 4; ++ky) {
        int iy = min(max(iy0 + ky, 0), IMG - 1);
        float rs = 0.f;
        #pragma unroll
        for (int kx = 0; kx < 4; ++kx) {
            int ix = min(max(ix0 + kx, 0), IMG - 1);
            rs += cW4[kx] * in[iy * IMG + ix];
        }
        acc += cW4[ky] * rs;
    }
    g2[t] = acc;
}

// ---------------- bicubic downsample, factor 4: 72 -> 18 ----------------
__global__ __launch_bounds__(256) void down4_kernel(const float* __restrict__ gt,
                                                    float* __restrict__ g4) {
    int t = blockIdx.x * 256 + threadIdx.x;
    if (t >= NB * 3 * 18 * 18) return;
    int bc = t / 324;
    int p  = t % 324;
    int oy = p / 18, ox = p % 18;
    const float* in = gt + (size_t)bc * (IMG * IMG);
    int iy0 = 4 * oy, ix0 = 4 * ox;   // frac 0.5 -> taps 4o..4o+3
    float acc = 0.f;
    #pragma unroll
    for (int ky = 0; ky < 4; ++ky) {
        int iy = min(max(iy0 + ky, 0), IMG - 1);
        float rs = 0.f;
        #pragma unroll
        for (int kx = 0; kx < 4; ++kx) {
            int ix = min(max(ix0 + kx, 0), IMG - 1);
            rs += cW4[kx] * in[iy * IMG + ix];
        }
        acc += cW4[ky] * rs;
    }
    g4[t] = acc;
}

// ---------------- 3x3x3 patch gram descriptor (dim 9) ----------------
__device__ __forceinline__ void gram27(const float* __restrict__ img, int dim,
                                       int pi, int pj, float* __restrict__ desc) {
    float ps[3][9];
    int hw = dim * dim;
    #pragma unroll
    for (int c = 0; c < 3; ++c)
        #pragma unroll
        for (int e = 0; e < 9; ++e)
            ps[c][e] = img[c * hw + (pi * 3 + (e / 3)) * dim + (pj * 3 + (e % 3))];
    const float inv27 = 1.0f / 27.0f;
    #pragma unroll
    for (int c = 0; c < 3; ++c)
        #pragma unroll
        for (int d = 0; d < 3; ++d) {
            float s = 0.f;
            #pragma unroll
            for (int e = 0; e < 9; ++e) s += ps[c][e] * ps[d][e];
            desc[c * 3 + d] = s * inv27;
        }
}

// one thread per descriptor: p1 (from x) and padded p2_cat (from gt, g2, g4)
__global__ __launch_bounds__(256) void grams_kernel(const float* __restrict__ x,
                                                    const float* __restrict__ gt,
                                                    const float* __restrict__ g2,
                                                    const float* __restrict__ g4,
                                                    float* __restrict__ p1,
                                                    float* __restrict__ pcat) {
    int t = blockIdx.x * 256 + threadIdx.x;
    if (t < NB * N1) {
        int b = t / N1, n = t % N1;
        float desc[9];
        gram27(x + (size_t)b * 3 * IMG * IMG, IMG, n / 24, n % 24, desc);
        float* o = p1 + (size_t)t * 9;
        #pragma unroll
        for (int k = 0; k < 9; ++k) o[k] = desc[k];
    } else if (t < NB * N1 + NB * MPAD) {
        int t2 = t - NB * N1;
        int b = t2 / MPAD, r = t2 % MPAD;
        float desc[9];
        if (r < 576) {
            gram27(gt + (size_t)b * 3 * IMG * IMG, IMG, r / 24, r % 24, desc);
        } else if (r < 720) {
            int pn = r - 576;
            gram27(g2 + (size_t)b * 3 * 36 * 36, 36, pn / 12, pn % 12, desc);
        } else if (r < 756) {
            int pn = r - 720;
            gram27(g4 + (size_t)b * 3 * 18 * 18, 18, pn / 6, pn % 6, desc);
        } else {
            #pragma unroll
            for (int k = 0; k < 9; ++k) desc[k] = 0.f;
        }
        float nrm = 0.f;
        #pragma unroll
        for (int k = 0; k < 9; ++k) nrm += desc[k] * desc[k];
        if (r >= MCAT) nrm = 1.0e30f;   // padded columns can never win argmin
        float* o = pcat + (size_t)t2 * RSTR;
        #pragma unroll
        for (int k = 0; k < 9; ++k) o[k] = desc[k];
        o[9]  = 0.f; o[10] = 0.f; o[11] = 0.f;
        o[12] = nrm; o[13] = 0.f; o[14] = 0.f; o[15] = 0.f;
    }
}

// ---------------- WMMA score + argmin ----------------
// argmin_m [ ||p1n-pm||^2 + ||p2n-pm||^2 ] == argmin_m [ ||pm||^2 - (p1n+p2n).pm ]
// A = -(p1+p2) (16x4 f32 fragments), C initialized to ||pm||^2, so D is the score.
// One wave per (batch, 16-row tile); K=9 padded to 12 -> 3 chained WMMAs per tile.
__global__ __launch_bounds__(128) void score_argmin_kernel(const float* __restrict__ p1,
                                                           const float* __restrict__ pcat,
                                                           int* __restrict__ idx) {
    int wid   = blockIdx.x * 4 + (threadIdx.x >> 5);   // 36 blocks * 4 waves = 144
    int lane  = threadIdx.x & 31;
    int b     = wid / 36;
    int rt    = wid % 36;
    int lhalf = lane >> 4;          // 0: lanes 0-15, 1: lanes 16-31
    int l15   = lane & 15;

    // A operand (16x4 f32 layout): lane<16 holds K={k0,k0+1}, lane>=16 K={k0+2,k0+3}
    int row = rt * 16 + l15;
    const float* p1r = p1   + ((size_t)b * N1   + row) * 9;
    const float* p2r = pcat + ((size_t)b * MPAD + row) * RSTR;
    v2f a0, a1, a2;
    {
        int kb = 2 * lhalf;
        float q0 = p1r[kb]     + p2r[kb];
        float q1 = p1r[kb + 1] + p2r[kb + 1];
        float q4 = p1r[kb + 4] + p2r[kb + 4];
        float q5 = p1r[kb + 5] + p2r[kb + 5];
        float q8 = (kb + 8 < 9) ? (p1r[kb + 8] + p2r[kb + 8]) : 0.f;
        float q9 = 0.f;                     // k = 9..11 are zero padding
        v2f t0 = {-q0, -q1}; a0 = t0;       // negate so D = ||pm||^2 - q.pm
        v2f t1 = {-q4, -q5}; a1 = t1;
        v2f t2 = {-q8, -q9}; a2 = t2;
    }

    const float* pcb = pcat + (size_t)b * MPAD * RSTR;

    float best[8];
    int   bidx[8];
    #pragma unroll
    for (int v = 0; v < 8; ++v) { best[v] = 3.0e38f; bidx[v] = 0x7fffffff; }

    #pragma unroll 4
    for (int j = 0; j < MPAD / 16; ++j) {
        int m = j * 16 + l15;                       // column this lane covers
        const float* pm = pcb + (size_t)m * RSTR;   // 64B-aligned padded row
        // issue all independent loads first: one clause, one wait, then 3 WMMAs
        v2f bv0 = *(const v2f*)(pm + 0 + 2 * lhalf);
        v2f bv1 = *(const v2f*)(pm + 4 + 2 * lhalf);
        v2f bv2 = *(const v2f*)(pm + 8 + 2 * lhalf);
        float nm = pm[12];
        v8f c = {nm, nm, nm, nm, nm, nm, nm, nm};   // C = ||pm||^2 (same col for all v)
        c = __builtin_amdgcn_wmma_f32_16x16x4_f32(false, a0, false, bv0, (short)0, c, false, false);
        c = __builtin_amdgcn_wmma_f32_16x16x4_f32(false, a1, false, bv1, (short)0, c, false, false);
        c = __builtin_amdgcn_wmma_f32_16x16x4_f32(false, a2, false, bv2, (short)0, c, false, false);
        // strict < keeps the first (smallest m) index within this slot automatically,
        // since m is strictly increasing over j.
        #pragma unroll
        for (int v = 0; v < 8; ++v) {               // c[v]: M = v + 8*lhalf, N = l15
            if (c[v] < best[v]) { best[v] = c[v]; bidx[v] = m; }
        }
    }

    // reduce the 16 columns of each half-tile across lanes (wave32, width 16);
    // here different lanes hold different m, so tie-break on smaller index.
    #pragma unroll
    for (int mask = 1; mask < 16; mask <<= 1) {
        #pragma unroll
        for (int v = 0; v < 8; ++v) {
            float os = __shfl_xor(best[v], mask, 16);
            int   oi = __shfl_xor(bidx[v], mask, 16);
            if (os < best[v] || (os == best[v] && oi < bidx[v])) { best[v] = os; bidx[v] = oi; }
        }
    }

    if (l15 == 0) {                         // lane 0 -> rows rt*16+0..7, lane 16 -> +8..15
        int base = b * N1 + rt * 16 + lhalf * 8;
        #pragma unroll
        for (int v = 0; v < 8; ++v) idx[base + v] = bidx[v];
    }
}

// ---------------- deterministic L1 mean reduction ----------------
__global__ __launch_bounds__(256) void l1_partial_kernel(const float* __restrict__ p1,
                                                         const float* __restrict__ pcat,
                                                         const int* __restrict__ idx,
                                                         float* __restrict__ part) {
    __shared__ float sdata[256];
    int t = blockIdx.x * 256 + threadIdx.x;
    float s = 0.f;
    if (t < NB * N1) {
        int b = t / N1;
        int m = idx[t];
        const float* pa = p1 + (size_t)t * 9;
        const float* pb = pcat + ((size_t)b * MPAD + m) * RSTR;
        #pragma unroll
        for (int d = 0; d < 9; ++d) s += fabsf(pa[d] - pb[d]);
    }
    sdata[threadIdx.x] = s;
    __syncthreads();
    for (int off = 128; off > 0; off >>= 1) {
        if (threadIdx.x < off) sdata[threadIdx.x] += sdata[threadIdx.x + off];
        __syncthreads();
    }
    if (threadIdx.x == 0) part[blockIdx.x] = sdata[0];
}

__global__ void l1_final_kernel(const float* __restrict__ part, float* __restrict__ out) {
    if (threadIdx.x == 0) {
        float s = 0.f;
        for (int i = 0; i < 9; ++i) s += part[i];
        out[0] = s / TOTEL;
    }
}

// ---------------- launch ----------------
extern "C" void kernel_launch(void* const* d_in, const int* in_sizes, int n_in,
                              void* d_out, int out_size, void* d_ws, size_t ws_size,
                              hipStream_t stream) {
    const float* x  = (const float*)d_in[0];
    const float* gt = (const float*)d_in[1];
    float* ws   = (float*)d_ws;
    float* p1   = ws + OFF_P1;
    float* pcat = ws + OFF_PCAT;
    float* g2   = ws + OFF_G2;
    float* g4   = ws + OFF_G4;
    int*   idx  = (int*)(ws + OFF_IDX);
    float* part = ws + OFF_PART;
    float* out  = (float*)d_out;

    down2_kernel<<<(NB * 3 * 36 * 36 + 255) / 256, 256, 0, stream>>>(gt, g2);
    down4_kernel<<<(NB * 3 * 18 * 18 + 255) / 256, 256, 0, stream>>>(gt, g4);
    grams_kernel<<<(NB * N1 + NB * MPAD + 255) / 256, 256, 0, stream>>>(
        x, gt, g2, g4, p1, pcat);
    score_argmin_kernel<<<36, 128, 0, stream>>>(p1, pcat, idx);
    l1_partial_kernel<<<9, 256, 0, stream>>>(p1, pcat, idx, part);
    l1_final_kernel<<<1, 32, 0, stream>>>(part, out);
}